// QuantumLayer_MultiQunode_48438641164570
// MI455X (gfx1250) — compile-verified
//
#include <hip/hip_runtime.h>
#include <hip/hip_bf16.h>

// ---------------------------------------------------------------------------
// CV quantum circuit simulator for MI455X (gfx1250, wave32, WMMA).
//   CUTOFF=14, M=3 modes -> state dim 2744 complex64, BATCH=1024, 2 circuits.
// Phase 1: build gates (expm) in workspace; beam-splitter 196x196 complex expm
//          uses a workgroup-cooperative complex GEMM on v_wmma_f32_16x16x4_f32.
// Phase 2: evolve all 2048 states; psi lives in LDS (double buffered, 44KB),
//          two-mode gates applied with WMMA f32 (U4 streamed from L2).
// Workspace requirement: ~29 MB (gates + expm scratch), all fits in 192MB L2.
// ---------------------------------------------------------------------------

#define CUT   14
#define D2    196        // 14^2
#define DIM   2744       // 14^3
#define G196  38416      // 196^2
#define NL    2
#define NC    2
#define M3    3
#define KP    3
#define BATCHN 1024
#define EXPM_S 12
#define EXPM_TERMS 14

typedef __attribute__((ext_vector_type(2))) float v2f;
typedef __attribute__((ext_vector_type(8))) float v8f;

// workspace layout (units: float)
static constexpr size_t BSU_OFF = 0;                                   // 24 * 2*G196  : BS unitaries (re|im)
static constexpr size_t BSS_OFF = BSU_OFF + (size_t)24 * 2 * G196;     // 24 * 3*2*G196: expm scratch (B,term,tmp)
static constexpr size_t SM_OFF  = BSS_OFF + (size_t)24 * 3 * 2 * G196; // 24 * 392     : squeeze/disp 14x14 gates
static constexpr size_t ID_OFF  = SM_OFF  + (size_t)24 * 2 * D2;       // 6144 * 14    : D(x)|0> columns

__device__ __forceinline__ v8f wmma4(v2f a, v2f b, v8f c) {
  // D(16x16,f32) = A(16x4,f32) x B(4x16,f32) + C
  return __builtin_amdgcn_wmma_f32_16x16x4_f32(false, a, false, b, (short)0, c,
                                               false, false);
}

// ---------------------------------------------------------------------------
// Workgroup-cooperative complex GEMM, C = scale * (A @ B), all 196x196,
// row-major re/im planes in global memory. 8 waves x (13x13) 16x16 tiles.
// Lane guards only predicate loads/stores; every WMMA runs with full EXEC.
// ---------------------------------------------------------------------------
__device__ void cgemm196(const float* __restrict__ Ar, const float* __restrict__ Ai,
                         const float* __restrict__ Br, const float* __restrict__ Bi,
                         float* __restrict__ Cr, float* __restrict__ Ci, float scale) {
  const int tid  = threadIdx.x;
  const int wave = tid >> 5;
  const int lane = tid & 31;
  const int lo   = lane & 15;   // M (A rows / D cols) index within tile
  const int hi   = lane >> 4;   // K half-select

  for (int tile = wave; tile < 13 * 13; tile += 8) {
    const int tm = tile / 13, tn = tile % 13;
    const int row  = tm * 16 + lo;   // A row
    const int coln = tn * 16 + lo;   // B col
    const bool rok = row  < D2;
    const bool cok = coln < D2;
    v8f acc_rr = {}, acc_ii = {}, acc_ri = {};
    for (int k0 = 0; k0 < D2; k0 += 4) {
      const int ka = k0 + 2 * hi;               // this lane's K pair: ka, ka+1
      v2f ar = {}, ai = {}, br = {}, bi = {};
      if (rok) {
        ar.x = Ar[row * D2 + ka]; ar.y = Ar[row * D2 + ka + 1];
        ai.x = Ai[row * D2 + ka]; ai.y = Ai[row * D2 + ka + 1];
      }
      if (cok) {
        br.x = Br[ka * D2 + coln]; br.y = Br[(ka + 1) * D2 + coln];
        bi.x = Bi[ka * D2 + coln]; bi.y = Bi[(ka + 1) * D2 + coln];
      }
      acc_rr = wmma4(ar, br, acc_rr);
      acc_ii = wmma4(ai, bi, acc_ii);
      acc_ri = wmma4(ar, bi, acc_ri);
      acc_ri = wmma4(ai, br, acc_ri);
    }
    for (int v = 0; v < 8; ++v) {
      const int r = tm * 16 + v + 8 * hi;
      const int c = tn * 16 + lo;
      if (r < D2 && c < D2) {
        Cr[r * D2 + c] = scale * (acc_rr[v] - acc_ii[v]);
        Ci[r * D2 + c] = scale * acc_ri[v];
      }
    }
  }
}

// ---------------------------------------------------------------------------
// Beam-splitter gates: build B = H * 2^-12 from sparse kron structure, then
// expm by 14 Taylor terms + 12 squarings. One workgroup per gate (24 total).
// ---------------------------------------------------------------------------
__global__ void __launch_bounds__(256)
bs_expm_kernel(float* __restrict__ ws,
               const float* __restrict__ th1, const float* __restrict__ ph1,
               const float* __restrict__ th2, const float* __restrict__ ph2) {
  const int g    = blockIdx.x;          // g = (cl*2 + slot)*3 + p
  const int p    = g % KP;
  const int s3   = g / KP;
  const int slot = s3 & 1;
  const int cl   = s3 >> 1;             // ci*NL + l
  const int pidx = cl * KP + p;
  const float theta = slot ? th2[pidx] : th1[pidx];
  const float phi   = slot ? ph2[pidx] : ph1[pidx];

  float* Ur = ws + BSU_OFF + (size_t)g * (2 * G196);
  float* Ui = Ur + G196;
  float* scr = ws + BSS_OFF + (size_t)g * (3 * 2 * G196);
  float* Br = scr;               float* Bi = Br + G196;
  float* Tr = scr + 2 * G196;    float* Ti = Tr + G196;
  float* Xr = scr + 4 * G196;    float* Xi = Xr + G196;

  const int tid = threadIdx.x;
  const float sc = 1.0f / 4096.0f;                  // 2^-12
  const float ct = theta * cosf(phi) * sc;
  const float st = theta * sinf(phi) * sc;

  // H = theta*(e^{i phi} kron(AD,A) - e^{-i phi} kron(A,AD))
  for (int e = tid; e < G196; e += 256) {
    const int I = e / D2, J = e % D2;
    const int i1 = I / CUT, i2 = I % CUT, j1 = J / CUT, j2 = J % CUT;
    float a1v = 0.f, a2v = 0.f;
    if ((i1 == j1 + 1) && (j2 == i2 + 1)) a1v = sqrtf((float)i1) * sqrtf((float)j2);
    if ((j1 == i1 + 1) && (i2 == j2 + 1)) a2v = sqrtf((float)j1) * sqrtf((float)i2);
    Br[e] = ct * (a1v - a2v);
    Bi[e] = st * (a1v + a2v);
    const float id = (I == J) ? 1.f : 0.f;
    Tr[e] = id; Ti[e] = 0.f; Ur[e] = id; Ui[e] = 0.f;
  }
  __syncthreads();

  for (int k = 1; k <= EXPM_TERMS; ++k) {
    cgemm196(Tr, Ti, Br, Bi, Xr, Xi, 1.0f / (float)k);   // term = term @ (B/k)
    __syncthreads();
    for (int e = tid; e < G196; e += 256) {
      const float xr = Xr[e], xi = Xi[e];
      Tr[e] = xr; Ti[e] = xi; Ur[e] += xr; Ui[e] += xi;  // out += term
    }
    __syncthreads();
  }
  for (int s = 0; s < EXPM_S; ++s) {
    cgemm196(Ur, Ui, Ur, Ui, Xr, Xi, 1.0f);              // out = out @ out
    __syncthreads();
    for (int e = tid; e < G196; e += 256) { Ur[e] = Xr[e]; Ui[e] = Xi[e]; }
    __syncthreads();
  }
}

// ---------------------------------------------------------------------------
// Squeeze / displacement gates: 14x14 complex expm entirely in LDS.
// One workgroup per gate (24 total); 196 active threads, one per element.
// ---------------------------------------------------------------------------
__global__ void __launch_bounds__(256)
sm_gates_kernel(float* __restrict__ ws,
                const float* __restrict__ rr, const float* __restrict__ phr,
                const float* __restrict__ aa, const float* __restrict__ pha) {
  __shared__ float Br[D2], Bi[D2], Tr[D2], Ti[D2], Or[D2], Oi[D2];
  const int g    = blockIdx.x;         // g = (cl*2 + type)*3 + m
  const int m    = g % 3;
  const int t3   = g / 3;
  const int type = t3 & 1;             // 0=squeeze, 1=disp
  const int cl   = t3 >> 1;
  const int pidx = cl * 3 + m;
  const int tid  = threadIdx.x;
  const float sc = 1.0f / 4096.0f;

  if (tid < D2) {
    const int i = tid / CUT, j = tid % CUT;
    float hr = 0.f, hi = 0.f;
    if (type == 0) {                    // H = 0.5*(conj(z)A^2 - z AD^2)
      const float rv = rr[pidx], ph = phr[pidx];
      const float zr = rv * cosf(ph), zi = rv * sinf(ph);
      if (j == i + 2) { const float c = 0.5f * sqrtf((float)((i + 1) * (i + 2))); hr =  c * zr; hi = -c * zi; }
      if (i == j + 2) { const float c = 0.5f * sqrtf((float)((j + 1) * (j + 2))); hr = -c * zr; hi = -c * zi; }
    } else {                            // H = alpha*AD - conj(alpha)*A
      const float av = aa[pidx], ph = pha[pidx];
      const float ar = av * cosf(ph), ai = av * sinf(ph);
      if (i == j + 1) { const float c = sqrtf((float)i); hr =  c * ar; hi =  c * ai; }
      if (j == i + 1) { const float c = sqrtf((float)j); hr = -c * ar; hi =  c * ai; }
    }
    Br[tid] = hr * sc; Bi[tid] = hi * sc;
    const float id = (i == j) ? 1.f : 0.f;
    Tr[tid] = id; Ti[tid] = 0.f; Or[tid] = id; Oi[tid] = 0.f;
  }
  __syncthreads();

  for (int k = 1; k <= EXPM_TERMS; ++k) {
    float nr = 0.f, ni = 0.f;
    if (tid < D2) {
      const int i = tid / CUT, j = tid % CUT;
      const float inv = 1.f / (float)k;
      for (int l = 0; l < CUT; ++l) {
        const float tr = Tr[i * CUT + l], ti = Ti[i * CUT + l];
        const float br = Br[l * CUT + j] * inv, bi = Bi[l * CUT + j] * inv;
        nr += tr * br - ti * bi;
        ni += tr * bi + ti * br;
      }
    }
    __syncthreads();
    if (tid < D2) { Tr[tid] = nr; Ti[tid] = ni; Or[tid] += nr; Oi[tid] += ni; }
    __syncthreads();
  }
  for (int s = 0; s < EXPM_S; ++s) {
    float nr = 0.f, ni = 0.f;
    if (tid < D2) {
      const int i = tid / CUT, j = tid % CUT;
      for (int l = 0; l < CUT; ++l) {
        const float xr = Or[i * CUT + l], xi = Oi[i * CUT + l];
        const float yr = Or[l * CUT + j], yi = Oi[l * CUT + j];
        nr += xr * yr - xi * yi;
        ni += xr * yi + xi * yr;
      }
    }
    __syncthreads();
    if (tid < D2) { Or[tid] = nr; Oi[tid] = ni; }
    __syncthreads();
  }
  if (tid < D2) {
    float* dst = ws + SM_OFF + (size_t)g * (2 * D2);
    dst[tid] = Or[tid]; dst[D2 + tid] = Oi[tid];
  }
}

// ---------------------------------------------------------------------------
// Initial displacement D(x)=exp(x*(AD-A)) (real). D hits |0>, so only column 0
// is needed. One workgroup per (b, ci, m) gate (6144 total).
// ---------------------------------------------------------------------------
__global__ void __launch_bounds__(256)
init_disp_kernel(float* __restrict__ ws, const float* __restrict__ x) {
  __shared__ float B[D2], T[D2], O[D2];
  const int gid = blockIdx.x;          // (b*NC + ci)*3 + m
  const int m   = gid % 3;
  const int bc  = gid / 3;
  const int ci  = bc % NC;
  const int b   = bc / NC;
  const float xv = x[b * (NC * M3) + ci * M3 + m];
  const int tid = threadIdx.x;

  if (tid < D2) {
    const int i = tid / CUT, j = tid % CUT;
    float h = 0.f;
    if (i == j + 1) h =  xv * sqrtf((float)i);
    if (j == i + 1) h = -xv * sqrtf((float)j);
    B[tid] = h * (1.0f / 4096.0f);
    const float id = (i == j) ? 1.f : 0.f;
    T[tid] = id; O[tid] = id;
  }
  __syncthreads();

  for (int k = 1; k <= EXPM_TERMS; ++k) {
    float nv = 0.f;
    if (tid < D2) {
      const int i = tid / CUT, j = tid % CUT;
      const float inv = 1.f / (float)k;
      for (int l = 0; l < CUT; ++l) nv += T[i * CUT + l] * B[l * CUT + j] * inv;
    }
    __syncthreads();
    if (tid < D2) { T[tid] = nv; O[tid] += nv; }
    __syncthreads();
  }
  for (int s = 0; s < EXPM_S; ++s) {
    float nv = 0.f;
    if (tid < D2) {
      const int i = tid / CUT, j = tid % CUT;
      for (int l = 0; l < CUT; ++l) nv += O[i * CUT + l] * O[l * CUT + j];
    }
    __syncthreads();
    if (tid < D2) O[tid] = nv;
    __syncthreads();
  }
  if (tid < CUT) ws[ID_OFF + (size_t)gid * CUT + tid] = O[tid * CUT + 0];
}

// ---------------------------------------------------------------------------
// Two-mode gate application via WMMA.
//   m1==0: out[J][c] = sum_L U[J][L] psi[L][c]       (196x196)x(196x14)
//   m1==1: out[a][J] = sum_L psi[a][L] U[J][L]       (14x196)x(196x196)^T
// psi in LDS; U4 streamed from global (L2-resident, shared by all 2048 WGs).
// ---------------------------------------------------------------------------
__device__ void apply_bs(const float* __restrict__ U4r, const float* __restrict__ U4i,
                         int m1, const float* pr, const float* pi,
                         float* qr, float* qi, int tid) {
  const int wave = tid >> 5, lane = tid & 31;
  const int lo = lane & 15, hi = lane >> 4;
  if (m1 == 0) {
    for (int tm = wave; tm < 13; tm += 8) {
      const int row = tm * 16 + lo;
      const bool rok = row < D2;
      if (rok) __builtin_prefetch(U4r + row * D2, 0, 1);
      v8f arr = {}, aii = {}, ari = {};
      for (int k0 = 0; k0 < D2; k0 += 4) {
        const int ka = k0 + 2 * hi;
        v2f ar = {}, ai = {}, br = {}, bi = {};
        if (rok) {
          ar.x = U4r[row * D2 + ka]; ar.y = U4r[row * D2 + ka + 1];
          ai.x = U4i[row * D2 + ka]; ai.y = U4i[row * D2 + ka + 1];
        }
        if (lo < CUT) {
          br.x = pr[ka * CUT + lo]; br.y = pr[(ka + 1) * CUT + lo];
          bi.x = pi[ka * CUT + lo]; bi.y = pi[(ka + 1) * CUT + lo];
        }
        arr = wmma4(ar, br, arr);
        aii = wmma4(ai, bi, aii);
        ari = wmma4(ar, bi, ari);
        ari = wmma4(ai, br, ari);
      }
      for (int v = 0; v < 8; ++v) {
        const int J = tm * 16 + v + 8 * hi;
        if (J < D2 && lo < CUT) {
          qr[J * CUT + lo] = arr[v] - aii[v];
          qi[J * CUT + lo] = ari[v];
        }
      }
    }
  } else {
    for (int tn = wave; tn < 13; tn += 8) {
      const int coln = tn * 16 + lo;
      const bool cok = coln < D2;
      if (cok) __builtin_prefetch(U4r + coln * D2, 0, 1);
      v8f arr = {}, aii = {}, ari = {};
      for (int k0 = 0; k0 < D2; k0 += 4) {
        const int ka = k0 + 2 * hi;
        v2f ar = {}, ai = {}, br = {}, bi = {};
        if (lo < CUT) {
          ar.x = pr[lo * D2 + ka]; ar.y = pr[lo * D2 + ka + 1];
          ai.x = pi[lo * D2 + ka]; ai.y = pi[lo * D2 + ka + 1];
        }
        if (cok) {
          br.x = U4r[coln * D2 + ka]; br.y = U4r[coln * D2 + ka + 1];
          bi.x = U4i[coln * D2 + ka]; bi.y = U4i[coln * D2 + ka + 1];
        }
        arr = wmma4(ar, br, arr);
        aii = wmma4(ai, bi, aii);
        ari = wmma4(ar, bi, ari);
        ari = wmma4(ai, br, ari);
      }
      for (int v = 0; v < 8; ++v) {
        const int a = v + 8 * hi;
        if (a < CUT && cok) {
          qr[a * D2 + coln] = arr[v] - aii[v];
          qi[a * D2 + coln] = ari[v];
        }
      }
    }
  }
}

// single-mode 14x14 gate along axis m (U cached in LDS)
__device__ void apply_sm(const float* gUr, const float* gUi, int m,
                         const float* pr, const float* pi,
                         float* qr, float* qi, int tid) {
  const int stride = (m == 0) ? D2 : ((m == 1) ? CUT : 1);
  for (int e = tid; e < DIM; e += 256) {
    const int a0 = e / D2, r2 = e % D2, a1 = r2 / CUT, a2 = r2 % CUT;
    const int am = (m == 0) ? a0 : ((m == 1) ? a1 : a2);
    const int base = e - am * stride;
    float accr = 0.f, acci = 0.f;
    for (int l = 0; l < CUT; ++l) {
      const float ur = gUr[am * CUT + l], ui = gUi[am * CUT + l];
      const float xr = pr[base + l * stride], xi = pi[base + l * stride];
      accr += ur * xr - ui * xi;
      acci += ur * xi + ui * xr;
    }
    qr[e] = accr; qi[e] = acci;
  }
}

// ---------------------------------------------------------------------------
// State evolution: one workgroup per (batch, circuit) = 2048 workgroups.
// ---------------------------------------------------------------------------
__global__ void __launch_bounds__(256)
evolve_kernel(const float* __restrict__ ws,
              const float* __restrict__ vph1, const float* __restrict__ vph2,
              const float* __restrict__ kerr, float* __restrict__ out) {
  __shared__ float s0r[DIM], s0i[DIM], s1r[DIM], s1i[DIM];
  __shared__ float gUr[D2], gUi[D2];
  __shared__ float red[256];

  const int tid = threadIdx.x;
  const int ci  = blockIdx.x & 1;
  const int b   = blockIdx.x >> 1;
  float *pr = s0r, *pi = s0i, *qr = s1r, *qi = s1i;

  // initial state: outer product of the three D(x)|0> columns
  const float* d0 = ws + ID_OFF + (size_t)((b * NC + ci) * 3 + 0) * CUT;
  const float* d1 = d0 + CUT;
  const float* d2 = d1 + CUT;
  for (int e = tid; e < DIM; e += 256) {
    const int a0 = e / D2, r2 = e % D2, a1 = r2 / CUT, a2 = r2 % CUT;
    pr[e] = d0[a0] * d1[a1] * d2[a2];
    pi[e] = 0.f;
  }
  __syncthreads();

  const int pair_m1[KP] = {0, 1, 0};

  for (int l = 0; l < NL; ++l) {
    const int cl = ci * NL + l;

    for (int slot = 0; slot < 2; ++slot) {
      // beam splitters
      for (int p = 0; p < KP; ++p) {
        const float* U = ws + BSU_OFF + (size_t)((cl * 2 + slot) * KP + p) * (2 * G196);
        apply_bs(U, U + G196, pair_m1[p], pr, pi, qr, qi, tid);
        __syncthreads();
        float* t;
        t = pr; pr = qr; qr = t;
        t = pi; pi = qi; qi = t;
      }
      // fused rotation phases across all three modes
      {
        const float* vp = (slot == 0) ? (vph1 + cl * 3) : (vph2 + cl * 3);
        const float f0 = vp[0], f1 = vp[1], f2 = vp[2];
        for (int e = tid; e < DIM; e += 256) {
          const int a0 = e / D2, r2 = e % D2, a1 = r2 / CUT, a2 = r2 % CUT;
          const float ang = f0 * a0 + f1 * a1 + f2 * a2;
          const float c = cosf(ang), s = sinf(ang);
          const float xr = pr[e], xi = pi[e];
          pr[e] = xr * c - xi * s;
          pi[e] = xr * s + xi * c;
        }
        __syncthreads();
      }
      // squeeze (slot 0) / displacement (slot 1): 14x14 complex gates
      for (int m = 0; m < M3; ++m) {
        const float* U = ws + SM_OFF + (size_t)((cl * 2 + slot) * 3 + m) * (2 * D2);
        if (tid < D2) { gUr[tid] = U[tid]; gUi[tid] = U[D2 + tid]; }
        __syncthreads();
        apply_sm(gUr, gUi, m, pr, pi, qr, qi, tid);
        __syncthreads();
        float* t;
        t = pr; pr = qr; qr = t;
        t = pi; pi = qi; qi = t;
      }
    }
    // Kerr phases, fused across modes
    {
      const float k0 = kerr[cl * 3 + 0], k1 = kerr[cl * 3 + 1], k2 = kerr[cl * 3 + 2];
      for (int e = tid; e < DIM; e += 256) {
        const int a0 = e / D2, r2 = e % D2, a1 = r2 / CUT, a2 = r2 % CUT;
        const float ang = k0 * (float)(a0 * a0) + k1 * (float)(a1 * a1) + k2 * (float)(a2 * a2);
        const float c = cosf(ang), s = sinf(ang);
        const float xr = pr[e], xi = pi[e];
        pr[e] = xr * c - xi * s;
        pi[e] = xr * s + xi * c;
      }
      __syncthreads();
    }
  }

  // expectation values <psi| X_m |psi>, X tridiagonal with sqrt(n) off-diag
  for (int m = 0; m < M3; ++m) {
    const int stride = (m == 0) ? D2 : ((m == 1) ? CUT : 1);
    float part = 0.f;
    for (int e = tid; e < DIM; e += 256) {
      const int a0 = e / D2, r2 = e % D2, a1 = r2 / CUT, a2 = r2 % CUT;
      const int am = (m == 0) ? a0 : ((m == 1) ? a1 : a2);
      const int base = e - am * stride;
      float xr = 0.f, xi = 0.f;
      if (am > 0) {
        const float c = sqrtf((float)am);
        xr += c * pr[base + (am - 1) * stride];
        xi += c * pi[base + (am - 1) * stride];
      }
      if (am < CUT - 1) {
        const float c = sqrtf((float)(am + 1));
        xr += c * pr[base + (am + 1) * stride];
        xi += c * pi[base + (am + 1) * stride];
      }
      part += pr[e] * xr + pi[e] * xi;
    }
    red[tid] = part;
    __syncthreads();
    for (int s = 128; s > 0; s >>= 1) {
      if (tid < s) red[tid] += red[tid + s];
      __syncthreads();
    }
    if (tid == 0) out[b * (NC * M3) + ci * M3 + m] = red[0];
    __syncthreads();
  }
}

// ---------------------------------------------------------------------------
extern "C" void kernel_launch(void* const* d_in, const int* in_sizes, int n_in,
                              void* d_out, int out_size, void* d_ws, size_t ws_size,
                              hipStream_t stream) {
  (void)in_sizes; (void)n_in; (void)out_size; (void)ws_size;
  const float* x    = (const float*)d_in[0];
  const float* th1  = (const float*)d_in[1];
  const float* ph1  = (const float*)d_in[2];
  const float* vph1 = (const float*)d_in[3];
  const float* rr   = (const float*)d_in[4];
  const float* phr  = (const float*)d_in[5];
  const float* th2  = (const float*)d_in[6];
  const float* ph2  = (const float*)d_in[7];
  const float* vph2 = (const float*)d_in[8];
  const float* aa   = (const float*)d_in[9];
  const float* pha  = (const float*)d_in[10];
  const float* kk   = (const float*)d_in[11];
  float* ws  = (float*)d_ws;
  float* out = (float*)d_out;

  bs_expm_kernel<<<NC * NL * 2 * KP, 256, 0, stream>>>(ws, th1, ph1, th2, ph2);
  sm_gates_kernel<<<NC * NL * 2 * M3, 256, 0, stream>>>(ws, rr, phr, aa, pha);
  init_disp_kernel<<<BATCHN * NC * M3, 256, 0, stream>>>(ws, x);
  evolve_kernel<<<BATCHN * NC, 256, 0, stream>>>(ws, vph1, vph2, kk, out);
}